// ValueLayerSlow_26147760898097
// MI455X (gfx1250) — compile-verified
//
#include <hip/hip_runtime.h>
#include <hip/hip_bf16.h>
#include <math.h>

typedef float v2f __attribute__((ext_vector_type(2)));
typedef float v8f __attribute__((ext_vector_type(8)));
typedef unsigned int u32x4 __attribute__((ext_vector_type(4)));
typedef int i32x4 __attribute__((ext_vector_type(4)));
typedef int i32x8 __attribute__((ext_vector_type(8)));

#define NWIRES 10
#define QDIM   1024
#define MTOT   16384   // B*T = 64*256

#if defined(__HIP_DEVICE_COMPILE__) && __has_builtin(__builtin_amdgcn_tensor_load_to_lds)
#define HAVE_TDM 1
#else
#define HAVE_TDM 0
#endif

// ---------------------------------------------------------------------------
// Kernel 0: zero the output (accumulated with atomics across N-blocks)
// ---------------------------------------------------------------------------
__global__ void zero_out_k(float* __restrict__ out, int n) {
    int i = blockIdx.x * blockDim.x + threadIdx.x;
    if (i < n) out[i] = 0.0f;
}

// ---------------------------------------------------------------------------
// Kernel 1: build U^T, k-pair interleaved.
// Row k of Ut = circuit applied to basis state e_k (one workgroup per k).
// Storage: element (k, i) at Up[(k>>1)*2048 + 2*i + (k&1)], so the {K,K+1}
// values a WMMA B-fragment register pair needs are adjacent in memory -> one
// aligned 8B ds_load per fragment after the TDM copies pair-rows into LDS.
// Wire 0 is the MOST significant qubit: bit index b = 9 - wire.
// ---------------------------------------------------------------------------
__global__ __launch_bounds__(512)
void build_ut_k(const float* __restrict__ rx0, const float* __restrict__ ry0,
                const float* __restrict__ ry1,
                float* __restrict__ Upr, float* __restrict__ Upi) {
    __shared__ float sr[QDIM];
    __shared__ float si[QDIM];
    const int k   = blockIdx.x;
    const int tid = threadIdx.x;   // 0..511 : one amplitude pair per gate

    for (int i = tid; i < QDIM; i += 512) {
        sr[i] = (i == k) ? 1.0f : 0.0f;
        si[i] = 0.0f;
    }
    __syncthreads();

    // Layer 1: RX(rx0[j]) then RY(ry0[j]) on each wire j
    for (int j = 0; j < NWIRES; ++j) {
        const int b   = 9 - j;
        const int msk = 1 << b;
        const int i0  = ((tid >> b) << (b + 1)) | (tid & (msk - 1));
        const int i1  = i0 | msk;
        // RX: [[c, -i s], [-i s, c]]
        {
            const float c = cosf(rx0[j] * 0.5f);
            const float s = sinf(rx0[j] * 0.5f);
            const float a0r = sr[i0], a0i = si[i0];
            const float a1r = sr[i1], a1i = si[i1];
            sr[i0] = c * a0r + s * a1i;  si[i0] = c * a0i - s * a1r;
            sr[i1] = c * a1r + s * a0i;  si[i1] = c * a1i - s * a0r;
        }
        __syncthreads();
        // RY: [[c, -s], [s, c]]
        {
            const float c = cosf(ry0[j] * 0.5f);
            const float s = sinf(ry0[j] * 0.5f);
            const float a0r = sr[i0], a0i = si[i0];
            const float a1r = sr[i1], a1i = si[i1];
            sr[i0] = c * a0r - s * a1r;  si[i0] = c * a0i - s * a1i;
            sr[i1] = s * a0r + c * a1r;  si[i1] = s * a0i + c * a1i;
        }
        __syncthreads();
    }

    // CNOT ring: control j, target (j+1)%10, applied sequentially
    for (int j = 0; j < NWIRES; ++j) {
        const int cb = 9 - j;
        const int tb = 9 - ((j + 1) % NWIRES);
        const int tm = 1 << tb;
        const int i0 = ((tid >> tb) << (tb + 1)) | (tid & (tm - 1));
        if ((i0 >> cb) & 1) {
            const int i1 = i0 | tm;
            const float tr = sr[i0], ti = si[i0];
            sr[i0] = sr[i1];  si[i0] = si[i1];
            sr[i1] = tr;      si[i1] = ti;
        }
        __syncthreads();
    }

    // Layer 2: RY(ry1[j])
    for (int j = 0; j < NWIRES; ++j) {
        const int b   = 9 - j;
        const int msk = 1 << b;
        const int i0  = ((tid >> b) << (b + 1)) | (tid & (msk - 1));
        const int i1  = i0 | msk;
        const float c = cosf(ry1[j] * 0.5f);
        const float s = sinf(ry1[j] * 0.5f);
        const float a0r = sr[i0], a0i = si[i0];
        const float a1r = sr[i1], a1i = si[i1];
        sr[i0] = c * a0r - s * a1r;  si[i0] = c * a0i - s * a1i;
        sr[i1] = s * a0r + c * a1r;  si[i1] = s * a0i + c * a1i;
        __syncthreads();
    }

    // k-pair interleaved store
    const size_t rb = (size_t)(k >> 1) * (2 * QDIM) + (k & 1);
    for (int i = tid; i < QDIM; i += 512) {
        Upr[rb + 2 * i] = sr[i];
        Upi[rb + 2 * i] = si[i];
    }
}

// ---------------------------------------------------------------------------
// Kernel 2: per-row inverse L2 norm of x (also warms x into L2)
// ---------------------------------------------------------------------------
__global__ __launch_bounds__(256)
void norms_k(const float* __restrict__ x, float* __restrict__ invn) {
    __shared__ float red[256];
    const int row = blockIdx.x;
    const float4 v = ((const float4*)&x[(size_t)row * QDIM])[threadIdx.x];
    red[threadIdx.x] = v.x * v.x + v.y * v.y + v.z * v.z + v.w * v.w;
    __syncthreads();
    for (int o = 128; o > 0; o >>= 1) {
        if (threadIdx.x < o) red[threadIdx.x] += red[threadIdx.x + o];
        __syncthreads();
    }
    if (threadIdx.x == 0) invn[row] = 1.0f / sqrtf(red[0]);
}

// ---------------------------------------------------------------------------
// Kernel 3: fused GEMM + expectation values, TDM double-buffered.
//   Amps(M=16384, N=1024) = X(M x K=1024, real) x Ut(K x N, complex)
//   out[m][j] += invn[m]^2 * sum_n (Ar^2 + Ai^2) * sign_j(n)
// (inv-norm applied to probabilities: amplitudes are linear in psi)
//
// Workgroup: 256 threads (8 waves), 128x128 output tile, K-blocked by 16.
// Waves 0/1/2 each issue one TENSOR_LOAD_TO_LDS (A / Br / Bi) for block i+1
// while everyone runs WMMA on block i; handoff = s_wait_tensorcnt + barrier.
// TDM pad feature creates the bank-conflict-avoiding LDS strides:
//   A rows:      16 DW + 2 DW pad  -> stride 18 (conflict-free b64 A frags)
//   B pair-rows: 256 DW + 32 DW pad -> stride 288 = 32 mod 64, so the two
//   16-lane halves (K{0,1} vs K{2,3}) hit disjoint bank ranges on b64 loads.
// ---------------------------------------------------------------------------
#define KBLK    16
#define ASTRIDE 18
#define BSTRIDE 288                       // floats per pair-row in LDS
#define OFF_INV2 0
#define OFF_A    128
#define A_BUF    (128 * ASTRIDE)          // 2304 floats
#define OFF_BR   (OFF_A + 2 * A_BUF)      // 4736
#define B_BUF    ((KBLK / 2) * BSTRIDE)   // 2304 floats
#define OFF_BI   (OFF_BR + 2 * B_BUF)     // 9344
#define SMEM_FLOATS (OFF_BI + 2 * B_BUF)  // 13952 floats = 55808 bytes

#if HAVE_TDM
// 2D TDM load: tile (tile_d0 contiguous x tile_d1 rows) of fp32, row stride
// stride0 (elements), linear into LDS with pad_amount+1 DWORDs appended every
// 2^(pad_interval+1) DWORDs. D# packing per CDNA5 ISA.
__device__ __forceinline__ void tdm_load_2d(unsigned lds_byte_addr,
                                            const void* gptr,
                                            unsigned tile_d0, unsigned tile_d1,
                                            unsigned stride0,
                                            unsigned pad_interval,
                                            unsigned pad_amount) {
    const unsigned long long ga = (unsigned long long)(uintptr_t)gptr;
    u32x4 g0;
    g0[0] = 1u;                                   // count=1, user descriptor
    g0[1] = lds_byte_addr;                        // lds_addr
    g0[2] = (unsigned)ga;                         // global_addr[31:0]
    g0[3] = (unsigned)(ga >> 32) | (2u << 30);    // global_addr[56:32] | type=2
    i32x8 g1;
    g1[0] = (int)((2u << 16)                      // data_size = 4B
                  | (1u << 20)                    // pad_enable
                  | (pad_interval << 22) | (pad_amount << 25));
    g1[1] = 0;                                    // abar=0, tensor_dim0[15:0]=0
    g1[2] = (int)0x10u;                           // tensor_dim0=1<<20, dim1 lo=0
    g1[3] = (int)(0x10u | (tile_d0 << 16));       // tensor_dim1=1<<20 | tile_dim0
    g1[4] = (int)tile_d1;                         // tile_dim1, tile_dim2=0
    g1[5] = (int)stride0;                         // tensor_dim0_stride[31:0]
    g1[6] = 0;
    g1[7] = 0;
    const i32x4 z4 = {0, 0, 0, 0};                // 2D: groups 2/3 unused
    const i32x8 z8 = {0, 0, 0, 0, 0, 0, 0, 0};
    __builtin_amdgcn_tensor_load_to_lds(g0, g1, z4, z4, z8, 0);
}
#endif

__global__ __launch_bounds__(256)
void qgemm_k(const float* __restrict__ x, const float* __restrict__ invn,
             const float* __restrict__ Upr, const float* __restrict__ Upi,
             float* __restrict__ out) {
    extern __shared__ float smem[];

    const int tid  = threadIdx.x;
    const int lane = tid & 31;
    const int wave = tid >> 5;
    const int wm   = wave & 3;   // row sub-block (32 rows)
    const int wn   = wave >> 2;  // col sub-block (64 cols)
    const int mbase = blockIdx.y * 128;
    const int nbase = blockIdx.x * 128;
    const int l15   = lane & 15;
    const int lhalf = lane >> 4;

    // inv-norm^2 table for this row block
    if (tid < 128) {
        const float iv = invn[mbase + tid];
        smem[OFF_INV2 + tid] = iv * iv;
    }

    v8f accr[2][4], acci[2][4];
    const v8f vzero = {0.f, 0.f, 0.f, 0.f, 0.f, 0.f, 0.f, 0.f};
#pragma unroll
    for (int mt = 0; mt < 2; ++mt)
#pragma unroll
        for (int nt = 0; nt < 4; ++nt) { accr[mt][nt] = vzero; acci[mt][nt] = vzero; }

#if HAVE_TDM
    const unsigned lds0 = (unsigned)(unsigned long long)(uintptr_t)smem;
    // prologue: stage block 0 into buffer 0
    if (wave == 0)
        tdm_load_2d(lds0 + OFF_A * 4, x + (size_t)mbase * QDIM,
                    KBLK, 128, QDIM, 3, 1);
    else if (wave == 1)
        tdm_load_2d(lds0 + OFF_BR * 4, Upr + 2 * nbase,
                    256, KBLK / 2, 2 * QDIM, 7, 31);
    else if (wave == 2)
        tdm_load_2d(lds0 + OFF_BI * 4, Upi + 2 * nbase,
                    256, KBLK / 2, 2 * QDIM, 7, 31);
#endif

    const int nblk = QDIM / KBLK;  // 64
    for (int ib = 0; ib < nblk; ++ib) {
        const int buf  = ib & 1;
        const int kb   = ib * KBLK;
        const bool more = (ib + 1) < nblk;

#if HAVE_TDM
        // issue block ib+1 into the other buffer, then wait for block ib
        if (wave == 0) {
            if (more) {
                tdm_load_2d(lds0 + (OFF_A + (buf ^ 1) * A_BUF) * 4,
                            x + (size_t)mbase * QDIM + kb + KBLK,
                            KBLK, 128, QDIM, 3, 1);
                __builtin_amdgcn_s_wait_tensorcnt((short)1);
            } else {
                __builtin_amdgcn_s_wait_tensorcnt((short)0);
            }
        } else if (wave == 1) {
            if (more) {
                tdm_load_2d(lds0 + (OFF_BR + (buf ^ 1) * B_BUF) * 4,
                            Upr + (size_t)((kb + KBLK) >> 1) * (2 * QDIM) + 2 * nbase,
                            256, KBLK / 2, 2 * QDIM, 7, 31);
                __builtin_amdgcn_s_wait_tensorcnt((short)1);
            } else {
                __builtin_amdgcn_s_wait_tensorcnt((short)0);
            }
        } else if (wave == 2) {
            if (more) {
                tdm_load_2d(lds0 + (OFF_BI + (buf ^ 1) * B_BUF) * 4,
                            Upi + (size_t)((kb + KBLK) >> 1) * (2 * QDIM) + 2 * nbase,
                            256, KBLK / 2, 2 * QDIM, 7, 31);
                __builtin_amdgcn_s_wait_tensorcnt((short)1);
            } else {
                __builtin_amdgcn_s_wait_tensorcnt((short)0);
            }
        }
        (void)more;
#else
        // fallback: synchronous cooperative staging of block ib
        {
            const int r  = tid >> 1;            // 0..127
            const int kh = (tid & 1) * 8;
            const float4* sa = (const float4*)&x[(size_t)(mbase + r) * QDIM + kb + kh];
#pragma unroll
            for (int q = 0; q < 2; ++q) {
                const float4 v = sa[q];
                float* d = &smem[OFF_A + buf * A_BUF + r * ASTRIDE + kh + q * 4];
                d[0] = v.x; d[1] = v.y; d[2] = v.z; d[3] = v.w;
            }
            const int kp = tid >> 5;            // 0..7 (pair-rows)
            const int c0 = (tid & 31) * 8;      // 0..248
            const size_t gofs = (size_t)((kb >> 1) + kp) * (2 * QDIM) + 2 * nbase + c0;
            const float4* s_r = (const float4*)&Upr[gofs];
            const float4* s_i = (const float4*)&Upi[gofs];
#pragma unroll
            for (int q = 0; q < 2; ++q) {
                const float4 vr = s_r[q];
                const float4 vi = s_i[q];
                float* dr = &smem[OFF_BR + buf * B_BUF + kp * BSTRIDE + c0 + q * 4];
                float* di = &smem[OFF_BI + buf * B_BUF + kp * BSTRIDE + c0 + q * 4];
                dr[0] = vr.x; dr[1] = vr.y; dr[2] = vr.z; dr[3] = vr.w;
                di[0] = vi.x; di[1] = vi.y; di[2] = vi.z; di[3] = vi.w;
            }
        }
#endif
        __syncthreads();

        // ---- WMMA on block ib: 4 k-steps x (2 mt x 4 nt x {re,im}) ----
        const float* As  = &smem[OFF_A + buf * A_BUF];
        const float* Brs = &smem[OFF_BR + buf * B_BUF];
        const float* Bis = &smem[OFF_BI + buf * B_BUF];
#pragma unroll
        for (int kk = 0; kk < KBLK; kk += 4) {
            // A fragment: 16x4 f32 -> lane holds M=lane&15, regs K{0,1}|{2,3}
            v2f af[2];
#pragma unroll
            for (int mt = 0; mt < 2; ++mt)
                af[mt] = *(const v2f*)&As[(wm * 32 + mt * 16 + l15) * ASTRIDE
                                          + kk + 2 * lhalf];
            // B fragment: 4x16 f32; pair-interleave -> one aligned b64 per frag
            const int kp = (kk >> 1) + lhalf;   // pair-row: K = kk + 2*lhalf
            v2f brf[4], bif[4];
#pragma unroll
            for (int nt = 0; nt < 4; ++nt) {
                const int c2 = (wn * 64 + nt * 16 + l15) * 2;
                brf[nt] = *(const v2f*)&Brs[kp * BSTRIDE + c2];
                bif[nt] = *(const v2f*)&Bis[kp * BSTRIDE + c2];
            }
#pragma unroll
            for (int mt = 0; mt < 2; ++mt)
#pragma unroll
                for (int nt = 0; nt < 4; ++nt) {
                    accr[mt][nt] = __builtin_amdgcn_wmma_f32_16x16x4_f32(
                        false, af[mt], false, brf[nt], (short)0, accr[mt][nt], false, false);
                    acci[mt][nt] = __builtin_amdgcn_wmma_f32_16x16x4_f32(
                        false, af[mt], false, bif[nt], (short)0, acci[mt][nt], false, false);
                }
        }
        __syncthreads();
    }

    // ---- epilogue: probs*invn^2 -> 10 signed sums -> global atomics ----
    // C/D layout: VGPR v, lanes 0-15: (M=v, N=lane); lanes 16-31: (M=v+8, N=lane-16)
#pragma unroll
    for (int mt = 0; mt < 2; ++mt) {
        const int rlb = wm * 32 + mt * 16 + (lhalf << 3);  // local row base
        float iv[8];
#pragma unroll
        for (int v = 0; v < 8; ++v) iv[v] = smem[OFF_INV2 + rlb + v];
#pragma unroll
        for (int nt = 0; nt < 4; ++nt) {
            const v8f cr = accr[mt][nt];
            const v8f ci = acci[mt][nt];
            v8f p;
#pragma unroll
            for (int v = 0; v < 8; ++v)
                p[v] = (cr[v] * cr[v] + ci[v] * ci[v]) * iv[v];

            const int n = nbase + wn * 64 + nt * 16 + l15;  // amplitude index

            // total prob sum over tile's 16 cols (n bits>=4 uniform per group)
            v8f P = p;
#pragma unroll
            for (int m = 1; m < 16; m <<= 1)
#pragma unroll
                for (int v = 0; v < 8; ++v) P[v] += __shfl_xor(P[v], m, 32);

            // lane-dependent sign bits 0..3 -> ops j=6..9
            v8f Q[4];
#pragma unroll
            for (int b = 0; b < 4; ++b) {
                const float s = ((n >> b) & 1) ? -1.0f : 1.0f;
#pragma unroll
                for (int v = 0; v < 8; ++v) Q[b][v] = p[v] * s;
#pragma unroll
                for (int m = 1; m < 16; m <<= 1)
#pragma unroll
                    for (int v = 0; v < 8; ++v) Q[b][v] += __shfl_xor(Q[b][v], m, 32);
            }

            if (l15 == 0) {  // lanes 0 and 16 each own 8 states
                const int rowb = mbase + rlb;
#pragma unroll
                for (int v = 0; v < 8; ++v) {
                    float* o = out + (size_t)(rowb + v) * NWIRES;
#pragma unroll
                    for (int j = 0; j < 6; ++j) {
                        const float s = ((n >> (9 - j)) & 1) ? -1.0f : 1.0f;
                        atomicAdd(o + j, s * P[v]);
                    }
#pragma unroll
                    for (int j = 6; j < 10; ++j)
                        atomicAdd(o + j, Q[9 - j][v]);
                }
            }
        }
    }
}

// ---------------------------------------------------------------------------
extern "C" void kernel_launch(void* const* d_in, const int* in_sizes, int n_in,
                              void* d_out, int out_size, void* d_ws, size_t ws_size,
                              hipStream_t stream) {
    const float* x   = (const float*)d_in[0];  // (64,256,1024) f32
    const float* rx0 = (const float*)d_in[1];  // (10,) f32
    const float* ry0 = (const float*)d_in[2];  // (10,) f32
    const float* ry1 = (const float*)d_in[3];  // (10,) f32
    float* out = (float*)d_out;                // (64,256,10) f32

    // workspace layout: invn (16384) | Upr (1M) | Upi (1M)  ~= 8.1 MB
    float* ws   = (float*)d_ws;
    float* invn = ws;
    float* Upr  = ws + MTOT;
    float* Upi  = Upr + (size_t)QDIM * QDIM;

    zero_out_k<<<(out_size + 255) / 256, 256, 0, stream>>>(out, out_size);
    build_ut_k<<<QDIM, 512, 0, stream>>>(rx0, ry0, ry1, Upr, Upi);
    norms_k<<<MTOT, 256, 0, stream>>>(x, invn);
    qgemm_k<<<dim3(QDIM / 128, MTOT / 128), 256, SMEM_FLOATS * 4, stream>>>(
        x, invn, Upr, Upi, out);
}